// GAT_5798205850076
// MI455X (gfx1250) — compile-verified
//
#include <hip/hip_runtime.h>
#include <hip/hip_bf16.h>
#include <math.h>

// ---------------------------------------------------------------------------
// GAT x4 + MLP head for MI455X (gfx1250, wave32).
//  - GEMMs via V_WMMA_F32_16X16X4_F32; B pre-packed to fragment order,
//    A tile staged in LDS (bank-padded), no predication in the K loop.
//  - segment softmax / aggregation via dst-CSR, one wave per node, no FP atomics
// ---------------------------------------------------------------------------

typedef __attribute__((ext_vector_type(2))) float v2f;
typedef __attribute__((ext_vector_type(8))) float v8f;

#define NEG_SLOPE 0.2f
#define FMAX_CH 8          // 8 * 32 lanes = up to 256 features per node row
#define FI_MAX  240

__device__ __forceinline__ float lrelu(float v) {
    return v > 0.0f ? v : NEG_SLOPE * v;
}

// ---------------------------------------------------------------------------
// Pack W[fi,fo] into per-(col-tile, k-step, lane) WMMA B-fragment order,
// zero-padding ragged columns. Layout: Bpack[((tn*ksteps + s)*32 + lane)*2 + {0,1}]
// B-frag (4x16 f32): lanes 0-15 rows {k,k+1} col=n0+lane; lanes 16-31 rows {k+2,k+3}.
// ---------------------------------------------------------------------------
__global__ void pack_B(const float* __restrict__ W, float* __restrict__ Bpack,
                       int fi, int fo, int tiles_x)
{
    int idx = blockIdx.x * blockDim.x + threadIdx.x;
    int ksteps = fi >> 2;
    int total = tiles_x * ksteps * 32;
    if (idx >= total) return;
    int lane = idx & 31;
    int s    = (idx >> 5) % ksteps;
    int tn   = (idx >> 5) / ksteps;
    int half = lane >> 4, l = lane & 15;
    int kk   = 4 * s + 2 * half;
    int col  = tn * 16 + l;
    float b0 = 0.0f, b1 = 0.0f;
    if (col < fo) {
        b0 = W[(size_t)kk       * fo + col];
        b1 = W[(size_t)(kk + 1) * fo + col];
    }
    Bpack[(size_t)idx * 2 + 0] = b0;
    Bpack[(size_t)idx * 2 + 1] = b1;
}

// ---------------------------------------------------------------------------
// Z[N,fo] = X[N,fi] @ W[fi,fo].  One block per 16-row tile; the 16 x fi
// activation rows are staged in LDS once (row stride fi+1 words -> odd word
// stride, conflict-free b64 reads), then the block's 8 waves sweep col tiles.
// A-frag (16x4 f32): lanes 0-15 hold M=lane, K={k,k+1}; lanes 16-31 K={k+2,k+3}.
// D (16x16 f32): VGPR r -> row m0+r (lanes 0-15) / m0+8+r (lanes 16-31).
// fi is a multiple of 4 (80/160/240/180); N a multiple of 16 (50000).
// ---------------------------------------------------------------------------
__global__ __launch_bounds__(256) void gemm_wmma_f32(
    const float* __restrict__ X, const float* __restrict__ Bpack,
    float* __restrict__ Z, int fi, int fo, int tiles_x)
{
    __shared__ float As[16 * (FI_MAX + 1)];
    const int sstride = fi + 1;
    const int m0 = blockIdx.x * 16;

    // cooperative stage of 16 rows into LDS (coalesced along fi)
    for (int r = 0; r < 16; ++r)
        for (int c = threadIdx.x; c < fi; c += 256)
            As[r * sstride + c] = X[(size_t)(m0 + r) * fi + c];
    __syncthreads();

    const int wave = threadIdx.x >> 5;
    const int lane = threadIdx.x & 31;
    const int half = lane >> 4;
    const int l    = lane & 15;
    const int ksteps = fi >> 2;
    const float* arow = As + l * sstride + 2 * half;

    for (int tn = wave; tn < tiles_x; tn += 8) {
        const float* bp = Bpack + ((size_t)tn * ksteps * 32 + lane) * 2;
        v8f acc = {};
        #pragma unroll 5
        for (int s = 0; s < ksteps; ++s) {
            v2f a, b;
            a.x = arow[4 * s];         // ds_load_b64
            a.y = arow[4 * s + 1];
            b.x = bp[0];               // coalesced global_load_b64
            b.y = bp[1];
            bp += 64;
            acc = __builtin_amdgcn_wmma_f32_16x16x4_f32(
                      false, a, false, b, (short)0, acc, false, false);
        }
        int bcol = tn * 16 + l;
        if (bcol < fo) {
            union { v8f v; float f[8]; } u; u.v = acc;
            int rbase = m0 + half * 8;
            #pragma unroll
            for (int r = 0; r < 8; ++r)
                Z[(size_t)(rbase + r) * fo + bcol] = u.f[r];
        }
    }
}

// ---------------------------------------------------------------------------
// el[n] = z[n,:].al ; er[n] = z[n,:].ar   (one wave per node)
// ---------------------------------------------------------------------------
__global__ __launch_bounds__(256) void node_dots(
    const float* __restrict__ z, const float* __restrict__ al,
    const float* __restrict__ ar, float* __restrict__ el,
    float* __restrict__ er, int n, int fo)
{
    int node = (int)((blockIdx.x * blockDim.x + threadIdx.x) >> 5);
    int lane = threadIdx.x & 31;
    if (node >= n) return;
    const float* zr = z + (size_t)node * fo;
    float sl = 0.0f, sr = 0.0f;
    for (int f = lane; f < fo; f += 32) {
        float v = zr[f];
        sl += v * al[f];
        sr += v * ar[f];
    }
    #pragma unroll
    for (int o = 16; o > 0; o >>= 1) {
        sl += __shfl_xor(sl, o, 32);
        sr += __shfl_xor(sr, o, 32);
    }
    if (lane == 0) { el[node] = sl; er[node] = sr; }
}

// ---------------------------------------------------------------------------
// CSR build: degree count -> exclusive scan -> scatter
// ---------------------------------------------------------------------------
__global__ void count_deg(const int* __restrict__ dst, int* __restrict__ deg, int ne)
{
    int j = blockIdx.x * blockDim.x + threadIdx.x;
    if (j < ne) atomicAdd(&deg[dst[j]], 1);
}

__global__ __launch_bounds__(1024) void scan_rowptr(
    const int* __restrict__ deg, int* __restrict__ rowptr,
    int* __restrict__ cursor, int n)
{
    __shared__ int part[1024];
    int t = threadIdx.x;
    int chunk = (n + 1023) >> 10;
    int b0 = t * chunk;
    int b1 = min(n, b0 + chunk);
    int s = 0;
    for (int i = b0; i < b1; ++i) s += deg[i];
    part[t] = s;
    __syncthreads();
    for (int o = 1; o < 1024; o <<= 1) {
        int v = (t >= o) ? part[t - o] : 0;
        __syncthreads();
        part[t] += v;
        __syncthreads();
    }
    int run = (t == 0) ? 0 : part[t - 1];
    for (int i = b0; i < b1; ++i) {
        rowptr[i] = run;
        cursor[i] = run;
        run += deg[i];
    }
    if (t == 1023) rowptr[n] = part[1023];
}

__global__ void scatter_eid(const int* __restrict__ dst, int* __restrict__ cursor,
                            int* __restrict__ eid, int ne)
{
    int j = blockIdx.x * blockDim.x + threadIdx.x;
    if (j < ne) {
        int p = atomicAdd(&cursor[dst[j]], 1);
        eid[p] = j;
    }
}

// ---------------------------------------------------------------------------
// Per-node softmax stats over incoming edges (one wave per node).
// ---------------------------------------------------------------------------
__global__ __launch_bounds__(256) void node_softmax_stats(
    const int* __restrict__ rowptr, const int* __restrict__ eid,
    const int* __restrict__ src, const float* __restrict__ el,
    const float* __restrict__ er, float* __restrict__ mbuf,
    float* __restrict__ dbuf, int n)
{
    int node = (int)((blockIdx.x * blockDim.x + threadIdx.x) >> 5);
    int lane = threadIdx.x & 31;
    if (node >= n) return;
    int beg = rowptr[node], end = rowptr[node + 1];
    float ern = er[node];

    float mx = -INFINITY;
    for (int i = beg + lane; i < end; i += 32) {
        float s = lrelu(el[src[eid[i]]] + ern);
        mx = fmaxf(mx, s);
    }
    #pragma unroll
    for (int o = 16; o > 0; o >>= 1) mx = fmaxf(mx, __shfl_xor(mx, o, 32));

    float sum = 0.0f;
    for (int i = beg + lane; i < end; i += 32) {
        float s = lrelu(el[src[eid[i]]] + ern);
        sum += expf(s - mx);
    }
    #pragma unroll
    for (int o = 16; o > 0; o >>= 1) sum += __shfl_xor(sum, o, 32);

    if (lane == 0) { mbuf[node] = mx; dbuf[node] = sum; }
}

// ---------------------------------------------------------------------------
// alpha[j] = exp(lrelu(el[src]+er[dst]) - m[dst]) / (denom==0 ? 1 : denom)
// ---------------------------------------------------------------------------
__global__ void edge_alpha(const int* __restrict__ src, const int* __restrict__ dst,
                           const float* __restrict__ el, const float* __restrict__ er,
                           const float* __restrict__ mbuf, const float* __restrict__ dbuf,
                           float* __restrict__ alpha, int ne)
{
    int j = blockIdx.x * blockDim.x + threadIdx.x;
    if (j >= ne) return;
    int d = dst[j];
    float s  = lrelu(el[src[j]] + er[d]);
    float a  = expf(s - mbuf[d]);
    float dn = dbuf[d];
    alpha[j] = a / (dn == 0.0f ? 1.0f : dn);
}

// ---------------------------------------------------------------------------
// out[n,:] = sum_{e in in(n)} z[src[e],:] * alpha[e] + b ; optional ReLU.
// One wave per node, features lane-strided (coalesced z row reads).
// ---------------------------------------------------------------------------
__global__ __launch_bounds__(256) void node_aggregate(
    const int* __restrict__ rowptr, const int* __restrict__ eid,
    const int* __restrict__ src, const float* __restrict__ z,
    const float* __restrict__ alpha, const float* __restrict__ bias,
    float* __restrict__ out, int n, int fo, int do_relu)
{
    int node = (int)((blockIdx.x * blockDim.x + threadIdx.x) >> 5);
    int lane = threadIdx.x & 31;
    if (node >= n) return;

    float acc[FMAX_CH];
    #pragma unroll
    for (int k = 0; k < FMAX_CH; ++k) acc[k] = 0.0f;

    int beg = rowptr[node], end = rowptr[node + 1];
    for (int i = beg; i < end; ++i) {
        int e = eid[i];
        float a = alpha[e];
        const float* zr = z + (size_t)src[e] * fo;
        #pragma unroll
        for (int k = 0; k < FMAX_CH; ++k) {
            int f = lane + 32 * k;
            if (f < fo) acc[k] += zr[f] * a;
        }
    }
    float* orow = out + (size_t)node * fo;
    #pragma unroll
    for (int k = 0; k < FMAX_CH; ++k) {
        int f = lane + 32 * k;
        if (f < fo) {
            float v = acc[k] + bias[f];
            if (do_relu) v = fmaxf(v, 0.0f);
            orow[f] = v;
        }
    }
}

// ---------------------------------------------------------------------------
// hg[f] = mean over nodes of x[:,f]   (one block per feature)
// ---------------------------------------------------------------------------
__global__ __launch_bounds__(256) void col_mean(const float* __restrict__ x,
                                                float* __restrict__ hg,
                                                int n, int fo)
{
    __shared__ float sm[256];
    int f = blockIdx.x;
    float s = 0.0f;
    for (int i = threadIdx.x; i < n; i += 256) s += x[(size_t)i * fo + f];
    sm[threadIdx.x] = s;
    __syncthreads();
    for (int o = 128; o > 0; o >>= 1) {
        if (threadIdx.x < (unsigned)o) sm[threadIdx.x] += sm[threadIdx.x + o];
        __syncthreads();
    }
    if (threadIdx.x == 0) hg[f] = sm[0] / (float)n;
}

// ---------------------------------------------------------------------------
// 120 -> 60 -> 30 -> 2 MLP + softmax, single block.
// ---------------------------------------------------------------------------
__global__ __launch_bounds__(64) void mlp_head(
    const float* __restrict__ hg,
    const float* __restrict__ lw1, const float* __restrict__ lb1,
    const float* __restrict__ lw2, const float* __restrict__ lb2,
    const float* __restrict__ lw3, const float* __restrict__ lb3,
    float* __restrict__ out)
{
    __shared__ float r1[64], r2[32], lg[2];
    int t = threadIdx.x;
    if (t < 60) {
        float s = lb1[t];
        for (int k = 0; k < 120; ++k) s += hg[k] * lw1[k * 60 + t];
        r1[t] = fmaxf(s, 0.0f);
    }
    __syncthreads();
    if (t < 30) {
        float s = lb2[t];
        for (int k = 0; k < 60; ++k) s += r1[k] * lw2[k * 30 + t];
        r2[t] = fmaxf(s, 0.0f);
    }
    __syncthreads();
    if (t < 2) {
        float s = lb3[t];
        for (int k = 0; k < 30; ++k) s += r2[k] * lw3[k * 2 + t];
        lg[t] = s;
    }
    __syncthreads();
    if (t == 0) {
        float mx = fmaxf(lg[0], lg[1]);
        float e0 = expf(lg[0] - mx), e1 = expf(lg[1] - mx);
        float inv = 1.0f / (e0 + e1);
        out[0] = e0 * inv;
        out[1] = e1 * inv;
    }
}

// ---------------------------------------------------------------------------

extern "C" void kernel_launch(void* const* d_in, const int* in_sizes, int n_in,
                              void* d_out, int out_size, void* d_ws, size_t ws_size,
                              hipStream_t stream) {
    (void)n_in; (void)out_size; (void)ws_size;

    const float* h   = (const float*)d_in[0];
    const int*   src = (const int*)d_in[1];
    const int*   dst = (const int*)d_in[2];
    const int N = in_sizes[0] / 80;
    const int E = in_sizes[1];

    static const int dims[5] = {80, 160, 240, 180, 120};
    const float *W[4], *al[4], *ar[4], *bb[4];
    for (int i = 0; i < 4; ++i) {
        W[i]  = (const float*)d_in[3 + 4 * i + 0];
        al[i] = (const float*)d_in[3 + 4 * i + 1];
        ar[i] = (const float*)d_in[3 + 4 * i + 2];
        bb[i] = (const float*)d_in[3 + 4 * i + 3];
    }
    const float* lw1 = (const float*)d_in[19]; const float* lb1 = (const float*)d_in[20];
    const float* lw2 = (const float*)d_in[21]; const float* lb2 = (const float*)d_in[22];
    const float* lw3 = (const float*)d_in[23]; const float* lb3 = (const float*)d_in[24];

    float* out = (float*)d_out;          // [0..1] = res softmax
    float* att = out + 2;                // [2..2+E) = layer-4 alpha

    // ---- workspace layout (fp32 words) ----
    float* z     = (float*)d_ws;                 // N*240
    float* xa    = z  + (size_t)N * 240;         // N*240
    float* xb    = xa + (size_t)N * 240;         // N*240
    float* el    = xb + (size_t)N * 240;         // N
    float* er    = el + N;                       // N
    float* mbuf  = er + N;                       // N
    float* dbuf  = mbuf + N;                     // N
    float* alpha = dbuf + N;                     // E
    float* hg    = alpha + E;                    // 128
    float* Bpack = hg + 128;                     // <= 15*240*16 = 57600
    int*   deg    = (int*)(Bpack + 57600);       // N
    int*   rowptr = deg + N;                     // N+1
    int*   cursor = rowptr + N + 1;              // N
    int*   eid    = cursor + N;                  // E

    const int TB = 256;
    int eblocks = (E + TB - 1) / TB;
    int nwave_blocks = (N * 32 + TB - 1) / TB;   // one wave per node

    // ---- build dst-CSR once (graph is identical for all 4 layers) ----
    hipMemsetAsync(deg, 0, (size_t)N * sizeof(int), stream);
    count_deg<<<eblocks, TB, 0, stream>>>(dst, deg, E);
    scan_rowptr<<<1, 1024, 0, stream>>>(deg, rowptr, cursor, N);
    scatter_eid<<<eblocks, TB, 0, stream>>>(dst, cursor, eid, E);

    // ---- 4 GAT layers ----
    const float* x = h;
    float* bufs[2] = {xa, xb};
    for (int i = 0; i < 4; ++i) {
        int fi = dims[i], fo = dims[i + 1];
        int tiles_x = (fo + 15) / 16;
        int ksteps  = fi >> 2;

        int ptotal  = tiles_x * ksteps * 32;
        pack_B<<<(ptotal + TB - 1) / TB, TB, 0, stream>>>(W[i], Bpack, fi, fo, tiles_x);
        gemm_wmma_f32<<<N / 16, TB, 0, stream>>>(x, Bpack, z, fi, fo, tiles_x);

        node_dots<<<nwave_blocks, TB, 0, stream>>>(z, al[i], ar[i], el, er, N, fo);
        node_softmax_stats<<<nwave_blocks, TB, 0, stream>>>(rowptr, eid, src,
                                                            el, er, mbuf, dbuf, N);
        float* aptr = (i == 3) ? att : alpha;
        edge_alpha<<<eblocks, TB, 0, stream>>>(src, dst, el, er, mbuf, dbuf, aptr, E);
        node_aggregate<<<nwave_blocks, TB, 0, stream>>>(rowptr, eid, src, z, aptr,
                                                        bb[i], bufs[i & 1], N, fo,
                                                        /*relu=*/(i < 3) ? 1 : 0);
        x = bufs[i & 1];
    }

    // ---- readout ----
    col_mean<<<120, 256, 0, stream>>>(x, hg, N, 120);
    mlp_head<<<1, 64, 0, stream>>>(hg, lw1, lb1, lw2, lb2, lw3, lb3, out);
}